// ConditionBlock_20761871909040
// MI455X (gfx1250) — compile-verified
//
#include <hip/hip_runtime.h>
#include <hip/hip_bf16.h>
#include <math.h>

typedef float v2f __attribute__((ext_vector_type(2)));
typedef float v8f __attribute__((ext_vector_type(8)));

#define B_ 2
#define T_ 256
#define P_ 88
#define C_ 128
#define ROWS_ (B_*P_*T_)   // 45056 = rows of the (BP,T) feature matrix (also B*T*P)

__device__ __forceinline__ float sigf(float x){ return 1.0f/(1.0f+expf(-x)); }

// ---------------------------------------------------------------------------
// Embedding + concat + (B,T,P,7) -> (B*P, T, 7) transpose
// ---------------------------------------------------------------------------
__global__ __launch_bounds__(256) void embed_k(
    const int* __restrict__ y, const int* __restrict__ v, const float* __restrict__ m,
    const float* __restrict__ ey, const float* __restrict__ ev,
    float* __restrict__ x0)
{
  int tid = blockIdx.x*256 + threadIdx.x;
  const int total = B_*T_*P_;
  if (tid >= total) return;
  int p = tid % P_;
  int t = (tid / P_) % T_;
  int b = tid / (P_*T_);
  int yi = y[tid];
  int vi = v[tid];
  float* o = x0 + (((size_t)b*P_ + p)*T_ + t)*7;
  o[0]=ey[yi*4+0]; o[1]=ey[yi*4+1]; o[2]=ey[yi*4+2]; o[3]=ey[yi*4+3];
  o[4]=ev[vi*2+0]; o[5]=ev[vi*2+1];
  o[6]=m[tid];
}

// ---------------------------------------------------------------------------
// C[m][n] = sum_k A[m][k]*W[n][k] + b1[n] + b2[n]   (fp32 WMMA 16x16x4)
// Block = 128 threads (4 waves). Block tile 64x64, wave tile 16x64.
// K staged 16 per chunk; 16 WMMA per barrier pair.
//  As: [row][k], row stride 18 (even -> b64-aligned frags, bank-conflict-free).
//  Bs: pair-interleaved: (k=kq*4+kp*2+w, col) at (kq*2+kp)*128 + (col^(kp<<4))*2 + w
//      so a lane's (k,k+1) fragment is one aligned ds_load_b64.
// Staging is software-pipelined: chunk c+1 is fetched to registers during the
// WMMA compute of chunk c, so global latency overlaps the matrix math.
// M, N must be multiples of 64 here (45056, {128,256,384}); K = 7 or 128.
// ---------------------------------------------------------------------------
__global__ __launch_bounds__(128) void gemm_bias_wmma(
    const float* __restrict__ A, const float* __restrict__ W,
    const float* __restrict__ b1, const float* __restrict__ b2,
    float* __restrict__ C, int M, int N, int K)
{
  __shared__ float As[64*18];   // 4.5 KB
  __shared__ float Bs[16*64];   // 4 KB
  const int m0 = blockIdx.x * 64;
  const int n0 = blockIdx.y * 64;
  const int tid  = threadIdx.x;
  const int wave = tid >> 5;
  const int lane = tid & 31;
  const int lane15 = lane & 15;
  const int khalf  = (lane >> 4) * 2;        // lanes 0-15 -> K{0,1}, lanes 16-31 -> K{2,3}
  const int kpf    = khalf >> 1;             // 0 / 1

  v8f zero = {0.f,0.f,0.f,0.f,0.f,0.f,0.f,0.f};
  v8f acc[4] = {zero, zero, zero, zero};

  float2 av[4], bv[4];
  // register-staging loader: pure loads, no LDS traffic
  auto load_chunk = [&](int k0) {
    const bool full = (k0 + 16 <= K);
    #pragma unroll
    for (int it = 0; it < 4; ++it) {
      int pidx = tid + it*128;               // [0,512)
      int kk2  = pidx & 7;
      int r    = pidx >> 3;
      const float* src = A + (size_t)(m0 + r) * K + (k0 + kk2*2);
      if (full) {
        av[it] = *(const float2*)src;
      } else {
        av[it].x = (k0 + kk2*2     < K) ? src[0] : 0.0f;
        av[it].y = (k0 + kk2*2 + 1 < K) ? src[1] : 0.0f;
      }
    }
    #pragma unroll
    for (int it = 0; it < 4; ++it) {
      int pidx = tid + it*128;               // [0,512)
      int col  = pidx & 63;
      int kp   = (pidx >> 6) & 1;
      int kq   = pidx >> 7;
      const float* src = W + (size_t)(n0 + col) * K + (k0 + kq*4 + kp*2);
      if (full) {
        bv[it] = *(const float2*)src;
      } else {
        bv[it].x = (k0 + kq*4 + kp*2     < K) ? src[0] : 0.0f;
        bv[it].y = (k0 + kq*4 + kp*2 + 1 < K) ? src[1] : 0.0f;
      }
    }
  };
  // pure LDS stores of the staged registers
  auto store_chunk = [&]() {
    #pragma unroll
    for (int it = 0; it < 4; ++it) {
      int pidx = tid + it*128;
      int kk2  = pidx & 7;
      int r    = pidx >> 3;
      *(float2*)&As[r*18 + kk2*2] = av[it];
    }
    #pragma unroll
    for (int it = 0; it < 4; ++it) {
      int pidx = tid + it*128;
      int col  = pidx & 63;
      int kp   = (pidx >> 6) & 1;
      int kq   = pidx >> 7;
      *(float2*)&Bs[(kq*2 + kp)*128 + ((col ^ (kp << 4)) << 1)] = bv[it];
    }
  };

  const int kc = (K + 15) >> 4;
  load_chunk(0);                              // prologue fetch
  for (int cch = 0; cch < kc; ++cch) {
    store_chunk();
    __syncthreads();
    if (cch + 1 < kc) load_chunk((cch + 1) * 16);   // prefetch next during compute

    const int arow = wave*16 + lane15;
    #pragma unroll
    for (int kq = 0; kq < 4; ++kq) {
      // A fragment: 16x4, ISA layout (VGPR0: K0|K2, VGPR1: K1|K3) -> one b64
      v2f a = *(const v2f*)&As[arow*18 + kq*4 + khalf];
      #pragma unroll
      for (int cb = 0; cb < 4; ++cb) {
        const int col = cb*16 + lane15;
        v2f bfr = *(const v2f*)&Bs[(kq*2 + kpf)*128 + ((col ^ (kpf << 4)) << 1)];
        acc[cb] = __builtin_amdgcn_wmma_f32_16x16x4_f32(
            false, a, false, bfr, (short)0, acc[cb], false, false);
      }
    }
    __syncthreads();
  }

  // epilogue: C/D layout — VGPR r: lanes0-15 M=r, lanes16-31 M=r+8
  const int rbase = m0 + wave*16 + ((lane >> 4) << 3);
  #pragma unroll
  for (int cb = 0; cb < 4; ++cb) {
    const int col = n0 + cb*16 + lane15;
    float bias = (b1 ? b1[col] : 0.0f) + (b2 ? b2[col] : 0.0f);
    #pragma unroll
    for (int r = 0; r < 8; ++r) {
      C[(size_t)(rbase + r) * N + col] = acc[cb][r] + bias;
    }
  }
}

// ---------------------------------------------------------------------------
// LSTM recurrence: one wave per sequence, 8 waves/WG sharing whh^T in LDS.
// pre (R,T,256) already holds x@wih^T + bih + bhh. Writes h into out[...,off+u].
// ---------------------------------------------------------------------------
__global__ __launch_bounds__(256) void lstm_rec(
    const float* __restrict__ pre, const float* __restrict__ whh,
    float* __restrict__ out, int T, int off, int reverse)
{
  __shared__ float Wt[64*256];    // Wt[k][j] = whh[j][k]
  __shared__ float hb[8][64];
  __shared__ float gb[8][256];
  const int tid  = threadIdx.x;
  const int wave = tid >> 5;
  const int lane = tid & 31;
  const int row  = blockIdx.x*8 + wave;

  for (int idx = tid; idx < 64*256; idx += 256) {
    int k = idx >> 8, j = idx & 255;
    Wt[idx] = whh[j*64 + k];
  }
  hb[wave][lane]      = 0.0f;
  hb[wave][lane + 32] = 0.0f;
  __syncthreads();

  float c0 = 0.0f, c1 = 0.0f;
  const int j0 = lane * 8;
  const float* prow = pre + (size_t)row * T * 256;
  float*       orow = out + (size_t)row * T * 128 + off;

  for (int s = 0; s < T; ++s) {
    const int t = reverse ? (T - 1 - s) : s;
    const float* pt = prow + (size_t)t * 256 + j0;
    float4 g0 = *(const float4*)(pt);
    float4 g1 = *(const float4*)(pt + 4);
    #pragma unroll 4
    for (int k = 0; k < 64; ++k) {
      const float hk = hb[wave][k];
      const float* wr = &Wt[k*256 + j0];
      float4 w0 = *(const float4*)(wr);
      float4 w1 = *(const float4*)(wr + 4);
      g0.x += hk*w0.x; g0.y += hk*w0.y; g0.z += hk*w0.z; g0.w += hk*w0.w;
      g1.x += hk*w1.x; g1.y += hk*w1.y; g1.z += hk*w1.z; g1.w += hk*w1.w;
    }
    *(float4*)&gb[wave][j0]     = g0;
    *(float4*)&gb[wave][j0 + 4] = g1;
    __syncthreads();

    {
      int u = lane;
      float ig = sigf (gb[wave][u]);
      float fg = sigf (gb[wave][64  + u]);
      float gg = tanhf(gb[wave][128 + u]);
      float og = sigf (gb[wave][192 + u]);
      c0 = fg*c0 + ig*gg;
      float h = og * tanhf(c0);
      hb[wave][u] = h;
      orow[(size_t)t*128 + u] = h;
    }
    {
      int u = lane + 32;
      float ig = sigf (gb[wave][u]);
      float fg = sigf (gb[wave][64  + u]);
      float gg = tanhf(gb[wave][128 + u]);
      float og = sigf (gb[wave][192 + u]);
      c1 = fg*c1 + ig*gg;
      float h = og * tanhf(c1);
      hb[wave][u] = h;
      orow[(size_t)t*128 + u] = h;
    }
    __syncthreads();
  }
}

// ---------------------------------------------------------------------------
// Neighborhood attention, gather path (2k < L). One wave per (n,l).
// qkv rows: [q(128) | k(128) | v(128)]. Lane handles 4 channels.
// ---------------------------------------------------------------------------
__global__ __launch_bounds__(256) void na_small(
    const float* __restrict__ qkv, const float* __restrict__ rpb,
    float* __restrict__ out, int NN, int L, float scale)
{
  const int wave = threadIdx.x >> 5;
  const int lane = threadIdx.x & 31;
  const int pos  = blockIdx.x*8 + wave;
  if (pos >= NN*L) return;
  const int n = pos / L;
  const int l = pos - n*L;
  const int d0 = lane*4;

  float4 q = *(const float4*)(qkv + (size_t)pos*384 + d0);
  int s = l - 3; if (s < 0) s = 0; if (s > L - 7) s = L - 7;

  const float* kbase = qkv + ((size_t)n*L + s)*384 + 128 + d0;
  float sc[7];
  #pragma unroll
  for (int t = 0; t < 7; ++t) {
    float4 kv = *(const float4*)(kbase + (size_t)t*384);
    float p = q.x*kv.x + q.y*kv.y + q.z*kv.z + q.w*kv.w;
    #pragma unroll
    for (int msk = 16; msk > 0; msk >>= 1) p += __shfl_xor(p, msk, 32);
    sc[t] = p*scale + rpb[s + t - l + 6];
  }
  float mx = sc[0];
  #pragma unroll
  for (int t = 1; t < 7; ++t) mx = fmaxf(mx, sc[t]);
  float a[7], sum = 0.0f;
  #pragma unroll
  for (int t = 0; t < 7; ++t) { a[t] = expf(sc[t] - mx); sum += a[t]; }
  const float inv = 1.0f / sum;

  const float* vbase = qkv + ((size_t)n*L + s)*384 + 256 + d0;
  float4 o = {0.f,0.f,0.f,0.f};
  #pragma unroll
  for (int t = 0; t < 7; ++t) {
    float4 vv = *(const float4*)(vbase + (size_t)t*384);
    float w = a[t]*inv;
    o.x += w*vv.x; o.y += w*vv.y; o.z += w*vv.z; o.w += w*vv.w;
  }
  *(float4*)(out + (size_t)pos*128 + d0) = o;
}

// ---------------------------------------------------------------------------
// Neighborhood attention, dense path (2k >= L). One wave per (n,i), online softmax.
// ---------------------------------------------------------------------------
__global__ __launch_bounds__(256) void na_full(
    const float* __restrict__ qkv, const float* __restrict__ rpb,
    float* __restrict__ out, int NN, int L, int kk, float scale)
{
  const int wave = threadIdx.x >> 5;
  const int lane = threadIdx.x & 31;
  const int pos  = blockIdx.x*8 + wave;
  if (pos >= NN*L) return;
  const int n = pos / L;
  const int i = pos - n*L;
  const int d0 = lane*4;

  float4 q = *(const float4*)(qkv + (size_t)pos*384 + d0);
  int s = i - kk/2; if (s < 0) s = 0; if (s > L - kk) s = L - kk;

  const float* kbase = qkv + (size_t)n*L*384 + 128 + d0;
  const float* vbase = qkv + (size_t)n*L*384 + 256 + d0;

  float mrun = -3.0e38f, lrun = 0.0f;
  float4 acc = {0.f,0.f,0.f,0.f};
  for (int j = 0; j < L; ++j) {
    if (j < s || j >= s + kk) continue;      // uniform per wave
    float4 kv = *(const float4*)(kbase + (size_t)j*384);
    float p = q.x*kv.x + q.y*kv.y + q.z*kv.z + q.w*kv.w;
    #pragma unroll
    for (int msk = 16; msk > 0; msk >>= 1) p += __shfl_xor(p, msk, 32);
    int bi = j - i + kk - 1;
    if (bi < 0) bi = 0; if (bi > 2*kk - 2) bi = 2*kk - 2;
    float sj = p*scale + rpb[bi];
    float mn = fmaxf(mrun, sj);
    float corr = expf(mrun - mn);
    float e    = expf(sj - mn);
    lrun = lrun*corr + e;
    float4 vv = *(const float4*)(vbase + (size_t)j*384);
    acc.x = acc.x*corr + e*vv.x;
    acc.y = acc.y*corr + e*vv.y;
    acc.z = acc.z*corr + e*vv.z;
    acc.w = acc.w*corr + e*vv.w;
    mrun = mn;
  }
  const float inv = 1.0f / lrun;
  float4 o = {acc.x*inv, acc.y*inv, acc.z*inv, acc.w*inv};
  *(float4*)(out + (size_t)pos*128 + d0) = o;
}

// ---------------------------------------------------------------------------
// dir==0: (B,P,T,C) -> (B,T,P,C);  dir==1: (B,T,P,C) -> (B,P,T,C).  C=128, float4.
// ---------------------------------------------------------------------------
__global__ __launch_bounds__(256) void transpose_pt(
    const float* __restrict__ in, float* __restrict__ out, int dir)
{
  size_t tid = (size_t)blockIdx.x*256 + threadIdx.x;
  const size_t total = (size_t)ROWS_ * 32;
  if (tid >= total) return;
  int c4 = (int)(tid & 31);
  size_t r = tid >> 5;          // output row
  size_t inrow;
  if (dir == 0) {               // out rows ordered (b,t,p)
    int p = (int)(r % P_);
    int t = (int)((r / P_) % T_);
    int b = (int)(r / ((size_t)P_ * T_));
    inrow = ((size_t)b*P_ + p)*T_ + t;
  } else {                      // out rows ordered (b,p,t)
    int t = (int)(r % T_);
    int p = (int)((r / T_) % P_);
    int b = (int)(r / ((size_t)T_ * P_));
    inrow = ((size_t)b*T_ + t)*P_ + p;
  }
  float4 val = *(const float4*)(in + inrow*128 + (size_t)c4*4);
  *(float4*)(out + r*128 + (size_t)c4*4) = val;
}

// (B,T,P,C) -> (B,C,T,P)
__global__ __launch_bounds__(256) void final_out_k(
    const float* __restrict__ in, float* __restrict__ out)
{
  size_t tid = (size_t)blockIdx.x*256 + threadIdx.x;
  const size_t total = (size_t)B_*C_*T_*P_;
  if (tid >= total) return;
  int p = (int)(tid % P_);
  int t = (int)((tid / P_) % T_);
  int c = (int)((tid / ((size_t)P_*T_)) % C_);
  int b = (int)(tid / ((size_t)P_*T_*C_));
  out[tid] = in[(((size_t)b*T_ + t)*P_ + p)*C_ + c];
}

// ---------------------------------------------------------------------------
extern "C" void kernel_launch(void* const* d_in, const int* in_sizes, int n_in,
                              void* d_out, int out_size, void* d_ws, size_t ws_size,
                              hipStream_t stream)
{
  (void)in_sizes; (void)n_in; (void)out_size; (void)ws_size;
  const int*   y     = (const int*)  d_in[0];
  const int*   v     = (const int*)  d_in[1];
  const float* mrat  = (const float*)d_in[2];
  const float* emb_y = (const float*)d_in[3];
  const float* emb_v = (const float*)d_in[4];

  float* ws  = (float*)d_ws;
  float* X0  = ws;                               // 45056*7
  float* PRE = X0  + (size_t)ROWS_*7;            // 45056*256
  float* XL0 = PRE + (size_t)ROWS_*256;          // 45056*128
  float* F1  = XL0 + (size_t)ROWS_*128;          // 45056*128
  float* QKV = F1  + (size_t)ROWS_*128;          // 45056*384
  float* ATT = QKV + (size_t)ROWS_*384;          // 45056*128
  float* F2  = ATT + (size_t)ROWS_*128;          // 45056*128

  embed_k<<<(ROWS_ + 255)/256, 256, 0, stream>>>(y, v, mrat, emb_y, emb_v, X0);

  // --- biLSTM (2 layers x 2 directions) ---
  auto lstm_dir = [&](const float* x, int K, int base, float* outbuf, int off, int rev) {
    const float* wih = (const float*)d_in[base + 0];
    const float* whh = (const float*)d_in[base + 1];
    const float* bih = (const float*)d_in[base + 2];
    const float* bhh = (const float*)d_in[base + 3];
    gemm_bias_wmma<<<dim3(ROWS_/64, 256/64), 128, 0, stream>>>(
        x, wih, bih, bhh, PRE, ROWS_, 256, K);
    lstm_rec<<<(B_*P_)/8, 256, 0, stream>>>(PRE, whh, outbuf, T_, off, rev);
  };
  lstm_dir(X0,  7,   5, XL0, 0,  0);   // l0 fwd
  lstm_dir(X0,  7,   9, XL0, 64, 1);   // l0 rev
  lstm_dir(XL0, 128, 13, F1,  0,  0);  // l1 fwd
  lstm_dir(XL0, 128, 17, F1,  64, 1);  // l1 rev

  const float SCALE = 0.08838834764831845f;   // 128^-0.5

  auto na_block = [&](const float* in, float* outb, int base, bool small_path,
                      int NN, int L, int k) {
    const float* qw  = (const float*)d_in[base + 0];
    const float* qb  = (const float*)d_in[base + 1];
    const float* pw  = (const float*)d_in[base + 2];
    const float* pb  = (const float*)d_in[base + 3];
    const float* rpb = (const float*)d_in[base + 4];
    gemm_bias_wmma<<<dim3(ROWS_/64, 384/64), 128, 0, stream>>>(
        in, qw, qb, nullptr, QKV, ROWS_, 384, 128);
    if (small_path)
      na_small<<<(ROWS_ + 7)/8, 256, 0, stream>>>(QKV, rpb, ATT, NN, L, SCALE);
    else
      na_full<<<(ROWS_ + 7)/8, 256, 0, stream>>>(QKV, rpb, ATT, NN, L, k, SCALE);
    gemm_bias_wmma<<<dim3(ROWS_/64, 128/64), 128, 0, stream>>>(
        ATT, pw, pb, nullptr, outb, ROWS_, 128, 128);
  };

  const int TRBLK = (int)(((size_t)ROWS_*32 + 255)/256);
  na_block(F1, F2, 21, true,  B_*P_, T_, 7);     // na1t on (B*P, T)
  transpose_pt<<<TRBLK, 256, 0, stream>>>(F2, F1, 0);
  na_block(F1, F2, 26, false, B_*T_, P_, 87);    // na1f on (B*T, P)
  transpose_pt<<<TRBLK, 256, 0, stream>>>(F2, F1, 1);
  na_block(F1, F2, 31, true,  B_*P_, T_, 7);     // na2t
  transpose_pt<<<TRBLK, 256, 0, stream>>>(F2, F1, 0);
  na_block(F1, F2, 36, false, B_*T_, P_, 87);    // na2f

  final_out_k<<<(int)(((size_t)B_*C_*T_*P_ + 255)/256), 256, 0, stream>>>(
      F2, (float*)d_out);
}